// PosMLPAttention_29953101923046
// MI455X (gfx1250) — compile-verified
//
#include <hip/hip_runtime.h>

// ---------------------------------------------------------------------------
// PosMLPAttention, fused flash-style, CDNA5 (gfx1250) WMMA fp32.
//
//   B=64, S=H*W=4096, D=256 (kdim=dim), 8 heads * hd=32, HID=16.
//
//   Pass 0 (prep):   Q = q@Wq;  qk[b,c,h] = scale * <Q[b,h,:], Wkv[c, h*32:..]>
//                    wg[b,:] = q@Wg + bg        (pos-MLP weights, 65 floats)
//   Pass 1 (fused):  per (b, s-chunk): stream mem[b] once; per 128-row tile:
//                    logits = mem_tile @ qk^T  (WMMA f32 16x16x4, N=16 padded heads)
//                    + pos-MLP bias; online softmax (running m, l per head);
//                    am += p^T @ mem_tile      (WMMA f32 16x16x4, N=256 cols)
//   Pass 2 (combine): merge 4 chunk-partials; vout = am@Wkv_V; out = vout@Wo.
// ---------------------------------------------------------------------------

typedef __attribute__((ext_vector_type(2))) float v2f;
typedef __attribute__((ext_vector_type(4))) float v4f;
typedef __attribute__((ext_vector_type(8))) float v8f;

#define NB     64      // batches
#define SPB    4096    // s positions per batch (H*W)
#define FD     256     // feature dim (kdim)
#define NCH    4       // s-chunks per batch
#define SC     (SPB / NCH)   // 1024 s per chunk
#define TILE   128     // s rows per tile iteration
#define PBS    132     // pBuf row stride (pad vs 64-bank LDS)

static __device__ __forceinline__ v8f wmma4(v2f a, v2f b, v8f c) {
  // V_WMMA_F32_16X16X4_F32 : D = A(16x4 f32) * B(4x16 f32) + C(16x16 f32)
  return __builtin_amdgcn_wmma_f32_16x16x4_f32(
      /*neg_a=*/false, a, /*neg_b=*/false, b,
      /*c_mod=*/(short)0, c, /*reuse_a=*/false, /*reuse_b=*/false);
}

// ---------------------------------------------------------------------------
// Pass 0: per-batch tiny GEMMs (Q, qk, gating weights). 64 WGs x 256 thr.
// ---------------------------------------------------------------------------
__global__ __launch_bounds__(256) void pmlp_prep(
    const float* __restrict__ queries, const float* __restrict__ Wq,
    const float* __restrict__ Wkv, const float* __restrict__ Wg,
    const float* __restrict__ bg, float* __restrict__ qkG,
    float* __restrict__ wgG)
{
  const int b   = blockIdx.x;
  const int tid = threadIdx.x;
  __shared__ float qS[FD];

  const float* qrow = queries + b * FD;
  {
    float s = 0.f;
    for (int i = 0; i < FD; ++i) s += qrow[i] * Wq[i * FD + tid];
    qS[tid] = s;
  }
  __syncthreads();

  const float scale = 0.17677669529663687f;  // 1/sqrt(32)
  for (int idx = tid; idx < SPB; idx += 256) {   // 4096 = 256 c * 16 h(pad)
    const int c = idx >> 4, h = idx & 15;
    float s = 0.f;
    if (h < 8) {
      const float* wk = Wkv + (size_t)c * 512 + h * 32;
      const float* qh = qS + h * 32;
      #pragma unroll
      for (int d = 0; d < 32; ++d) s += qh[d] * wk[d];
      s *= scale;
    }
    qkG[(size_t)b * SPB + idx] = s;   // stored [c][16]
  }

  if (tid < 65) {
    float s = bg[tid];
    for (int i = 0; i < FD; ++i) s += qrow[i] * Wg[i * 65 + tid];
    wgG[b * 80 + tid] = s;
  }
}

// ---------------------------------------------------------------------------
// Pass 1: fused logits + online-softmax + am accumulation.
// grid (64 batches, 4 chunks), 256 threads = 8 wave32.
// ---------------------------------------------------------------------------
__global__ __launch_bounds__(256) void pmlp_fused(
    const float* __restrict__ mem, const float* __restrict__ pos,
    const float* __restrict__ qkG, const float* __restrict__ wgG,
    float* __restrict__ mG, float* __restrict__ lG, float* __restrict__ amG)
{
  const int b      = blockIdx.x;
  const int ch     = blockIdx.y;
  const int tid    = threadIdx.x;
  const int lane   = tid & 31;
  const int wave   = tid >> 5;
  const int halfHi = (lane >= 16);
  const int n16    = lane & 15;

  // qk pair-interleaved: element (c,h) at (c>>1)*32 + h*2 + (c&1)
  __shared__ __align__(16) float qkS[2048 * 2];     // 16 KB
  __shared__ __align__(16) float pBuf[16 * PBS];    // p tile, [head][row]
  __shared__ float biasS[TILE];
  __shared__ float wgS[65];
  __shared__ float redBuf[8 * 32];
  __shared__ float mRunS[16], lRunS[16], sfS[16];

  for (int idx = tid; idx < SPB; idx += 256) {
    const int c = idx >> 4, h = idx & 15;
    qkS[(c >> 1) * 32 + h * 2 + (c & 1)] = qkG[(size_t)b * SPB + idx];
  }
  if (tid < 65) wgS[tid] = wgG[b * 80 + tid];
  if (tid < 16) { mRunS[tid] = -1e30f; lRunS[tid] = 0.f; }

  const float pcx = pos[b * 4 + 0], pcy = pos[b * 4 + 1];
  const float pbw = pos[b * 4 + 2], pbh = pos[b * 4 + 3];

  const float* memB = mem + (size_t)b * SPB * FD;
  const int sChunk = ch * SC;

  v8f acc0 = {};   // am cols [wave*32 .. +15]
  v8f acc1 = {};   // am cols [wave*32+16 .. +31]

  for (int t = 0; t < SC / TILE; ++t) {
    const int sTile = sChunk + t * TILE;
    __syncthreads();   // protects pBuf/redBuf/biasS vs previous iteration

    // prefetch next tile of mem into cache hierarchy (global_prefetch_b8)
    if (t + 1 < SC / TILE) {
      const float* nb = memB + (size_t)(sTile + TILE) * FD;
      #pragma unroll
      for (int q = 0; q < 4; ++q)
        __builtin_prefetch(nb + ((size_t)tid + q * 256) * 32, 0, 1);
    }

    // ---- positional-MLP bias for the 128 rows of this tile ----
    if (tid < TILE) {
      const int s = sTile + tid;
      const int yi = s >> 6, xi = s & 63;
      const float gx = (xi + 0.5f) * (1.0f / 64.0f);
      const float gy = (yi + 0.5f) * (1.0f / 64.0f);
      const float rx = (gx - pcx) / pbw;
      const float ry = (gy - pcy) / pbh;
      float acc = wgS[64];
      #pragma unroll
      for (int n = 0; n < 16; ++n) {
        float hv = rx * wgS[2 * n] + ry * wgS[2 * n + 1] + wgS[32 + n];
        hv = hv > 0.f ? hv : 0.f;
        acc += hv * wgS[48 + n];
      }
      biasS[tid] = acc;
    }
    __syncthreads();

    // ---- WMMA #1: logits(16 rows x 16 heads) = memRows(16x256) @ qk(256x16)
    v8f lg = {};
    const float* aRow = memB + (size_t)(sTile + wave * 16 + n16) * FD + (halfHi ? 2 : 0);
    const float* bQk  = qkS + n16 * 2 + (halfHi ? 32 : 0);
    #pragma unroll 8
    for (int k = 0; k < FD; k += 4) {
      const v2f a  = *(const v2f*)(aRow + k);
      const v2f bb = *(const v2f*)(bQk + k * 16);
      lg = wmma4(a, bb, lg);
    }

    // ---- bias add + per-head tile max (cross-wave reduce via LDS) ----
    const int rowBase = wave * 16 + (halfHi ? 8 : 0);
    float logv[8];
    float lmax = -1e30f;
    #pragma unroll
    for (int i = 0; i < 8; ++i) {
      const float v = lg[i] + biasS[rowBase + i];
      logv[i] = v;
      lmax = fmaxf(lmax, v);
    }
    redBuf[wave * 32 + lane] = lmax;
    __syncthreads();
    if (tid < 16) {
      float tm = -1e30f;
      #pragma unroll
      for (int w = 0; w < 8; ++w)
        tm = fmaxf(tm, fmaxf(redBuf[w * 32 + tid], redBuf[w * 32 + 16 + tid]));
      const float mOld = mRunS[tid];
      const float mNew = fmaxf(mOld, tm);
      sfS[tid]   = __expf(mOld - mNew);
      mRunS[tid] = mNew;
    }
    __syncthreads();

    // ---- p = exp(logit - m_new); stage p^T into LDS; per-head partial sums
    const float mNew = mRunS[n16];
    float pv[8];
    float psum = 0.f;
    #pragma unroll
    for (int i = 0; i < 8; ++i) { pv[i] = __expf(logv[i] - mNew); psum += pv[i]; }
    float* pw = pBuf + n16 * PBS + rowBase;
    *(v4f*)(pw)     = (v4f){pv[0], pv[1], pv[2], pv[3]};
    *(v4f*)(pw + 4) = (v4f){pv[4], pv[5], pv[6], pv[7]};
    redBuf[wave * 32 + lane] = psum;
    __syncthreads();
    if (tid < 16) {
      float ts = 0.f;
      #pragma unroll
      for (int w = 0; w < 8; ++w)
        ts += redBuf[w * 32 + tid] + redBuf[w * 32 + 16 + tid];
      lRunS[tid] = lRunS[tid] * sfS[tid] + ts;
    }

    // ---- WMMA #2: am(16 heads x 32 cols) += p^T(16x128) @ memTile(128x32)
    #pragma unroll
    for (int i = 0; i < 8; ++i) {
      const float f = sfS[i + (halfHi ? 8 : 0)];   // online-softmax rescale
      acc0[i] *= f;
      acc1[i] *= f;
    }
    const int   col0  = wave * 32;
    const float* pA    = pBuf + n16 * PBS + (halfHi ? 2 : 0);
    const float* bCol0 = memB + (size_t)sTile * FD + col0 + n16;
    const float* bCol1 = bCol0 + 16;
    #pragma unroll 4
    for (int k = 0; k < TILE; k += 4) {
      const v2f a = *(const v2f*)(pA + k);
      const int krow = k + (halfHi ? 2 : 0);
      v2f b0, b1;
      b0.x = bCol0[(size_t)krow * FD];
      b0.y = bCol0[(size_t)(krow + 1) * FD];
      b1.x = bCol1[(size_t)krow * FD];
      b1.y = bCol1[(size_t)(krow + 1) * FD];
      acc0 = wmma4(a, b0, acc0);
      acc1 = wmma4(a, b1, acc1);
    }
  }

  // ---- epilogue: store chunk partials (m, l, am) ----
  const size_t base = (size_t)(b * NCH + ch) * 16;
  if (tid < 16) {
    mG[base + tid] = mRunS[tid];
    lG[base + tid] = lRunS[tid];
  }
  const int headOff = halfHi ? 8 : 0;
  #pragma unroll
  for (int i = 0; i < 8; ++i) {
    const size_t rowb = (base + i + headOff) * FD + wave * 32 + n16;
    amG[rowb]      = acc0[i];
    amG[rowb + 16] = acc1[i];
  }
}

// ---------------------------------------------------------------------------
// Pass 2: merge chunk partials, project through Wkv_V and Wo. 64 WGs x 256.
// ---------------------------------------------------------------------------
__global__ __launch_bounds__(256) void pmlp_combine(
    const float* __restrict__ mG, const float* __restrict__ lG,
    const float* __restrict__ amG, const float* __restrict__ Wkv,
    const float* __restrict__ Wo, float* __restrict__ out)
{
  const int b   = blockIdx.x;
  const int tid = threadIdx.x;
  __shared__ float coefS[NCH * 8];
  __shared__ float amS[8 * FD];
  __shared__ float voutS[FD];

  if (tid < 8) {
    float m[NCH], l[NCH];
    float M = -1e30f;
    #pragma unroll
    for (int c = 0; c < NCH; ++c) {
      m[c] = mG[(size_t)(b * NCH + c) * 16 + tid];
      l[c] = lG[(size_t)(b * NCH + c) * 16 + tid];
      M = fmaxf(M, m[c]);
    }
    float L = 0.f;
    float e[NCH];
    #pragma unroll
    for (int c = 0; c < NCH; ++c) { e[c] = __expf(m[c] - M); L += l[c] * e[c]; }
    const float inv = 1.f / L;
    #pragma unroll
    for (int c = 0; c < NCH; ++c) coefS[c * 8 + tid] = e[c] * inv;
  }
  __syncthreads();

  for (int idx = tid; idx < 8 * FD; idx += 256) {
    const int h = idx >> 8, c = idx & 255;
    float s = 0.f;
    #pragma unroll
    for (int k = 0; k < NCH; ++k)
      s += amG[((size_t)(b * NCH + k) * 16 + h) * FD + c] * coefS[k * 8 + h];
    amS[idx] = s;
  }
  __syncthreads();

  {  // vout[j] = sum_c am[h][c] * Wkv[c][256+j],  h = j/32
    const int j = tid;
    const float* w = Wkv + 256 + j;           // stride 512
    const float* a = amS + (j >> 5) * FD;
    float s = 0.f;
    for (int c = 0; c < FD; ++c) s += a[c] * w[(size_t)c * 512];
    voutS[j] = s;
  }
  __syncthreads();

  {  // out[b][j] = vout @ Wo
    float s = 0.f;
    for (int i = 0; i < FD; ++i) s += voutS[i] * Wo[i * FD + tid];
    out[(size_t)b * FD + tid] = s;
  }
}

// ---------------------------------------------------------------------------
extern "C" void kernel_launch(void* const* d_in, const int* in_sizes, int n_in,
                              void* d_out, int out_size, void* d_ws, size_t ws_size,
                              hipStream_t stream) {
  const float* queries = (const float*)d_in[0];
  const float* memory  = (const float*)d_in[1];
  const float* pos     = (const float*)d_in[2];
  const float* Wq      = (const float*)d_in[3];
  const float* Wkv     = (const float*)d_in[4];
  const float* Wo      = (const float*)d_in[5];
  const float* Wg      = (const float*)d_in[6];
  const float* bg      = (const float*)d_in[7];
  float* out = (float*)d_out;

  // workspace layout (floats); total ~1.33M floats ~ 5.3 MB
  float* ws  = (float*)d_ws;
  float* qkG = ws;                     // 64*4096      = 262144
  float* wgG = qkG + 262144;           // 64*80        = 5120
  float* mG  = wgG + 5120;             // 64*4*16      = 4096
  float* lG  = mG + 4096;              // 64*4*16      = 4096
  float* amG = lG + 4096;              // 64*4*16*256  = 1048576

  pmlp_prep<<<NB, 256, 0, stream>>>(queries, Wq, Wkv, Wg, bg, qkG, wgG);
  dim3 grid(NB, NCH);
  pmlp_fused<<<grid, 256, 0, stream>>>(memory, pos, qkG, wgG, mG, lG, amG);
  pmlp_combine<<<NB, 256, 0, stream>>>(mG, lG, amG, Wkv, Wo, out);
}